// GCNMasker_36438502540143
// MI455X (gfx1250) — compile-verified
//
#include <hip/hip_runtime.h>

// ---------------------------------------------------------------------------
// GCN masker for MI455X (gfx1250, wave32).
//   h       = relu(x @ W1 + b1)          -> V_WMMA_F32_16X16X4_F32 (fp32 WMMA)
//   e       = h[row]·Wa + h[col]·Wb + be -> thread-per-edge, L2-resident h
//   scores  = segment sparsemax(e, row)  -> per-row bucket + rank sort in LDS
//   new_adj = top-K(32) threshold        -> scatter into pre-zeroed 256MB out
// Output layout: d_out = [ h (8192*128 f32) | new_adj (8192*8192 f32) ]
// ---------------------------------------------------------------------------

typedef float v2f __attribute__((ext_vector_type(2)));
typedef float v4f __attribute__((ext_vector_type(4)));
typedef float v8f __attribute__((ext_vector_type(8)));

#define NROWS  8192
#define F_IN   256
#define NHID   128
#define KTOP   32
#define CAP    256   // per-row edge bucket capacity (mean occupancy = 64)

// ---------------------------------------------------------------------------
// Kernel 0: zero the new_adj output region (256 MB -> the HBM roofline floor)
// with nontemporal 128-bit stores, and zero the per-row edge counters.
// ---------------------------------------------------------------------------
__global__ __launch_bounds__(256) void zero_kernel(v4f* __restrict__ adj4, size_t n4,
                                                   int* __restrict__ counts, int nCounts)
{
    size_t i      = (size_t)blockIdx.x * blockDim.x + threadIdx.x;
    size_t stride = (size_t)gridDim.x * blockDim.x;
    v4f z = {0.0f, 0.0f, 0.0f, 0.0f};
    for (size_t k = i; k < n4; k += stride)
        __builtin_nontemporal_store(z, &adj4[k]);
    for (size_t k = i; k < (size_t)nCounts; k += stride)
        counts[k] = 0;
}

// ---------------------------------------------------------------------------
// Kernel 1: h = relu(x @ W1 + b1) using fp32 WMMA.
// One wave per 16x16 output tile; 512 M-tiles x 8 N-tiles = 4096 waves.
// A fragment (16x4 f32, 2 VGPRs): lane m=lane%16, K = k0 + 2*(lane/16) + {0,1}.
// B fragment (4x16 f32, 2 VGPRs): lane n=lane%16, same K split (mirrored).
// C/D (16x16 f32, 8 VGPRs): vgpr r -> row m = r + 8*(lane/16), col n = lane%16.
// ---------------------------------------------------------------------------
__global__ __launch_bounds__(256) void gemm_h_kernel(const float* __restrict__ x,
                                                     const float* __restrict__ W1,
                                                     const float* __restrict__ b1,
                                                     float* __restrict__ h)
{
    const int wave  = threadIdx.x >> 5;
    const int lane  = threadIdx.x & 31;
    const int tile  = blockIdx.x * 8 + wave;   // 0..4095
    const int mBase = (tile >> 3) * 16;        // 0..8176
    const int nBase = (tile & 7) * 16;         // 0..112
    const int half  = lane >> 4;               // 0 or 1
    const int l16   = lane & 15;

    const float* __restrict__ xr = x + (size_t)(mBase + l16) * F_IN;  // A row
    const float* __restrict__ wc = W1 + nBase + l16;                  // B column

    v8f acc = {};
    #pragma unroll 4
    for (int k0 = 0; k0 < F_IN; k0 += 4) {
        const int ka = k0 + 2 * half;
        v2f a, b;
        a.x = xr[ka];
        a.y = xr[ka + 1];
        b.x = wc[(size_t)ka * NHID];
        b.y = wc[(size_t)(ka + 1) * NHID];
        acc = __builtin_amdgcn_wmma_f32_16x16x4_f32(
                  /*neg_a=*/false, a, /*neg_b=*/false, b,
                  /*c_mod=*/(short)0, acc, /*reuse_a=*/false, /*reuse_b=*/false);
    }

    const float bias = b1[nBase + l16];
    #pragma unroll
    for (int r = 0; r < 8; ++r) {
        float v = acc[r] + bias;
        v = v > 0.0f ? v : 0.0f;
        h[(size_t)(mBase + r + 8 * half) * NHID + (nBase + l16)] = v;
    }
}

// ---------------------------------------------------------------------------
// Kernel 2: per-edge attention logit + bucket by source row.
//   e[j] = h[row]·We[:128] + h[col]·We[128:] + be
// h is 4 MB -> L2 resident; float4 loads keep VMEM wide.
// ---------------------------------------------------------------------------
__global__ __launch_bounds__(256) void edge_kernel(const float* __restrict__ h,
                                                   const int* __restrict__ ei,
                                                   const float* __restrict__ We,
                                                   const float* __restrict__ be,
                                                   int* __restrict__ counts,
                                                   float* __restrict__ zbuf,
                                                   int* __restrict__ cbuf,
                                                   int E)
{
    const int j = blockIdx.x * blockDim.x + threadIdx.x;
    if (j >= E) return;
    const int r = ei[j];
    const int c = ei[E + j];

    const float4* __restrict__ hr = (const float4*)(h + (size_t)r * NHID);
    const float4* __restrict__ hc = (const float4*)(h + (size_t)c * NHID);
    const float4* __restrict__ wa = (const float4*)(We);
    const float4* __restrict__ wb = (const float4*)(We + NHID);

    float e = be[0];
    #pragma unroll 8
    for (int q = 0; q < NHID / 4; ++q) {
        float4 a = hr[q], b = hc[q], p = wa[q], s = wb[q];
        e += a.x * p.x + a.y * p.y + a.z * p.z + a.w * p.w;
        e += b.x * s.x + b.y * s.y + b.z * s.z + b.w * s.w;
    }

    const int pos = atomicAdd(&counts[r], 1);
    if (pos < CAP) {
        zbuf[(size_t)r * CAP + pos] = e;
        cbuf[(size_t)r * CAP + pos] = c;
    }
}

// ---------------------------------------------------------------------------
// Kernel 3: per-row sparsemax + top-K threshold + scatter.
// One 256-thread workgroup per row. Rank-sort (strict total order, tie by
// index) in LDS; serial sparsemax scan by thread 0 (<=256 iters); support
// size kmax == number of positive scores, so thresh = kmax>=32 ? score[31]:0
// and only surviving entries are scattered into the pre-zeroed output.
// ---------------------------------------------------------------------------
__global__ __launch_bounds__(256) void row_kernel(const int* __restrict__ counts,
                                                  const float* __restrict__ zbuf,
                                                  const int* __restrict__ cbuf,
                                                  float* __restrict__ adj)
{
    __shared__ float sz[CAP];
    __shared__ int   sc[CAP];
    __shared__ float oz[CAP];
    __shared__ int   oc[CAP];
    __shared__ float s_tau;
    __shared__ int   s_kmax;

    const int row = blockIdx.x;
    int c = counts[row];
    if (c > CAP) c = CAP;
    const int t = threadIdx.x;

    if (t < c) {
        sz[t] = zbuf[(size_t)row * CAP + t];
        sc[t] = cbuf[(size_t)row * CAP + t];
    }
    __syncthreads();

    if (t < c) {
        const float zi = sz[t];
        int rank = 0;
        for (int j = 0; j < c; ++j) {
            const float zj = sz[j];
            rank += (zj > zi) || (zj == zi && j < t);
        }
        oz[rank] = zi;
        oc[rank] = sc[t];
    }
    __syncthreads();

    if (t == 0) {
        float cs = 0.0f, csk = 0.0f;
        int kmax = 0;
        for (int j = 0; j < c; ++j) {
            const float z = oz[j];
            cs += z;
            if (1.0f + (float)(j + 1) * z > cs) { kmax = j + 1; csk = cs; }
        }
        if (kmax < 1) { kmax = 1; csk = (c > 0) ? oz[0] : 0.0f; }
        s_kmax = kmax;
        s_tau  = (csk - 1.0f) / (float)kmax;
    }
    __syncthreads();

    if (c == 0) return;
    const int   kmax = s_kmax;
    const float tau  = s_tau;

    float thresh = 0.0f;
    if (kmax >= KTOP) {
        const float s31 = oz[KTOP - 1] - tau;
        thresh = s31 > 0.0f ? s31 : 0.0f;
    }

    if (t < kmax && t < c) {
        const float s = oz[t] - tau;
        if (s > 0.0f && s >= thresh)
            adj[(size_t)row * NROWS + oc[t]] = s;
    }
}

// ---------------------------------------------------------------------------
// Launch: zero -> WMMA gemm -> edge buckets -> row sparsemax/top-K.
// Workspace: [ counts: 8192 i32 | zbuf: 8192*256 f32 | cbuf: 8192*256 i32 ]
//            = 32 KB + 8 MB + 8 MB.
// ---------------------------------------------------------------------------
extern "C" void kernel_launch(void* const* d_in, const int* in_sizes, int n_in,
                              void* d_out, int out_size, void* d_ws, size_t ws_size,
                              hipStream_t stream)
{
    const float* x  = (const float*)d_in[0];
    const int*   ei = (const int*)d_in[1];
    const float* W1 = (const float*)d_in[2];
    const float* b1 = (const float*)d_in[3];
    const float* We = (const float*)d_in[4];
    const float* be = (const float*)d_in[5];

    float* h   = (float*)d_out;                               // 8192*128
    float* adj = (float*)d_out + (size_t)NROWS * NHID;        // 8192*8192

    int*   counts = (int*)d_ws;
    float* zbuf   = (float*)((char*)d_ws + (size_t)NROWS * sizeof(int));
    int*   cbuf   = (int*)((char*)d_ws + (size_t)NROWS * sizeof(int)
                                       + (size_t)NROWS * CAP * sizeof(float));

    const int E = in_sizes[1] / 2;

    const size_t adj4 = (size_t)NROWS * NROWS / 4;
    zero_kernel<<<4096, 256, 0, stream>>>((v4f*)adj, adj4, counts, NROWS);

    gemm_h_kernel<<<(NROWS / 16) * (NHID / 16) / 8, 256, 0, stream>>>(x, W1, b1, h);

    edge_kernel<<<(E + 255) / 256, 256, 0, stream>>>(h, ei, We, be,
                                                     counts, zbuf, cbuf, E);

    row_kernel<<<NROWS, 256, 0, stream>>>(counts, zbuf, cbuf, adj);
}